// xLSTMBlock_2551210574036
// MI455X (gfx1250) — compile-verified
//
#include <hip/hip_runtime.h>
#include <hip/hip_bf16.h>
#include <math.h>

#define Bsz 256
#define Isz 512
#define Hsz 512

typedef __attribute__((ext_vector_type(2))) float v2f;
typedef __attribute__((ext_vector_type(4))) float v4f;
typedef __attribute__((ext_vector_type(8))) float v8f;

// ---------------------------------------------------------------------------
// Kernel 1: six fused projections via V_WMMA_F32_16X16X4_F32.
// Output conceptually [B=256, 6*H=3072]; one wave computes one 16x16 tile,
// looping K over I=512 in steps of 4. A = x[16 x 4], B = W^T[4 x 16].
// Epilogue applies bias + softplus/sigmoid for the gate matrices.
// ---------------------------------------------------------------------------
__global__ __launch_bounds__(256) void proj_kernel(
    const float* __restrict__ x,
    const float* __restrict__ Wq, const float* __restrict__ Wk,
    const float* __restrict__ Wv, const float* __restrict__ Wi,
    const float* __restrict__ Wf, const float* __restrict__ Wo,
    const float* __restrict__ bi, const float* __restrict__ bf,
    const float* __restrict__ bo,
    float* __restrict__ qo, float* __restrict__ ko, float* __restrict__ vo,
    float* __restrict__ io, float* __restrict__ fo, float* __restrict__ oo)
{
    const int t    = threadIdx.x;
    const int wv   = t >> 5;
    const int lane = t & 31;

    const int tile = blockIdx.x * 8 + wv;   // 0..3071 (16 row-tiles x 192 col-tiles)
    const int mt   = tile / 192;            // batch row tile, 0..15
    const int nt   = tile % 192;            // column tile over the 6 stacked W's
    const int id   = nt >> 5;               // which matrix: 0=q 1=k 2=v 3=i 4=f 5=o
    const int hn   = (nt & 31) << 4;        // h base inside that matrix

    const float* W = Wq;
    if      (id == 1) W = Wk;
    else if (id == 2) W = Wv;
    else if (id == 3) W = Wi;
    else if (id == 4) W = Wf;
    else if (id == 5) W = Wo;

    // ISA VGPR layout for 32-bit 16x4 A / 4x16 B (wave32):
    //   lane r = lane&15 holds row r; K pair base = 2*(lane>>4); v2f = contiguous.
    const int r    = lane & 15;
    const int koff = (lane >> 4) << 1;      // 0 or 2
    const float* ap = x + (size_t)(mt * 16 + r) * Isz + koff;
    const float* bp = W + (size_t)(hn + r) * Isz + koff;   // B[k][n] = W[hn+n][kk+k]

    v8f acc = {};
    #pragma unroll 4
    for (int kk = 0; kk < Isz; kk += 4) {
        v2f a = *(const v2f*)(ap + kk);
        v2f b = *(const v2f*)(bp + kk);
        acc = __builtin_amdgcn_wmma_f32_16x16x4_f32(
            /*neg_a=*/false, a, /*neg_b=*/false, b,
            /*c_mod=*/(short)0, acc, /*reuse_a=*/false, /*reuse_b=*/false);
    }

    float* dst = qo;
    const float* bias = nullptr;
    if      (id == 1) dst = ko;
    else if (id == 2) dst = vo;
    else if (id == 3) { dst = io; bias = bi; }
    else if (id == 4) { dst = fo; bias = bf; }
    else if (id == 5) { dst = oo; bias = bo; }

    // C/D layout: VGPR e -> M = e + 8*(lane>>4); N = lane&15.
    const int n     = lane & 15;
    const int h     = hn + n;
    const int mbase = mt * 16 + ((lane >> 4) << 3);

    #pragma unroll
    for (int e = 0; e < 8; ++e) {
        float val = acc[e];
        if (id >= 3) {
            val += bias[h];
            if (id == 3) {
                // softplus, numerically safe
                val = (val > 20.f) ? val : log1pf(__expf(val));
            } else {
                val = 1.f / (1.f + __expf(-val));
            }
        }
        dst[(size_t)(mbase + e) * Hsz + h] = val;
    }
}

// ---------------------------------------------------------------------------
// Kernel 2: per-batch-row sum(q), then n/m state update and readout denom.
// One block per batch row.
// ---------------------------------------------------------------------------
__global__ __launch_bounds__(256) void gates_kernel(
    const float* __restrict__ qb, const float* __restrict__ kb,
    const float* __restrict__ ib, const float* __restrict__ fb,
    const float* __restrict__ n_prev, const float* __restrict__ m_prev,
    float* __restrict__ n_out, float* __restrict__ m_out,
    float* __restrict__ denom)
{
    __shared__ float red[256];
    const int b = blockIdx.x, t = threadIdx.x;
    const int base = b * Hsz;

    red[t] = qb[base + t] + qb[base + t + 256];
    __syncthreads();
    for (int off = 128; off; off >>= 1) {
        if (t < off) red[t] += red[t + off];
        __syncthreads();
    }
    const float qs = red[0];

    #pragma unroll
    for (int rr = 0; rr < 2; ++rr) {
        const int idx = base + t + rr * 256;
        const float iv = ib[idx], kv = kb[idx], fv = fb[idx];
        const float nv = fv * n_prev[idx] + iv * kv;
        n_out[idx] = nv;
        m_out[idx] = fmaxf(fv * m_prev[idx], iv);
        denom[idx] = fmaxf(nv * qs, 1.0f);
    }
}

// ---------------------------------------------------------------------------
// Kernel 3: the bandwidth-bound pass. C = f*C_prev + (i*k)*v with the readout
// dot num = sum_j C[i,j]*q[j] fused in (saves a full 268MB re-read of C).
// The 537MB C stream is touched exactly once and exceeds the 192MB L2, so the
// C_prev load and C store use NON-TEMPORAL hints (TH=NT) to avoid evicting the
// reusable working set (x, W's, q/v rows). Block: 8 waves handle 8 consecutive
// i-rows of one batch b; q,v staged in LDS; b128 fully-coalesced traffic;
// wave32 shuffle reduction for the dot.
// ---------------------------------------------------------------------------
__global__ __launch_bounds__(256) void cell_kernel(
    const float* __restrict__ C_prev, const float* __restrict__ vb,
    const float* __restrict__ qb, const float* __restrict__ ib,
    const float* __restrict__ kb, const float* __restrict__ fb,
    float* __restrict__ C_out, float* __restrict__ num)
{
    __shared__ float q_s[Hsz];
    __shared__ float v_s[Hsz];

    const int b      = blockIdx.x >> 6;        // 64 blocks per batch row
    const int i_base = (blockIdx.x & 63) << 3; // 8 i-rows per block
    const int t      = threadIdx.x;
    const int base   = b * Hsz;

    q_s[t]       = qb[base + t];
    q_s[t + 256] = qb[base + t + 256];
    v_s[t]       = vb[base + t];
    v_s[t + 256] = vb[base + t + 256];
    __syncthreads();

    const int w = t >> 5, lane = t & 31;
    const int i = i_base + w;
    const int gidx = base + i;
    const float fv = fb[gidx];
    const float s1 = ib[gidx] * kb[gidx];

    const size_t rowoff = (size_t)b * Hsz * Hsz + (size_t)i * Hsz;
    const v4f* cp = (const v4f*)(C_prev + rowoff);
    v4f*       co = (v4f*)(C_out + rowoff);
    const v4f* q4 = (const v4f*)q_s;
    const v4f* v4 = (const v4f*)v_s;

    float dot = 0.f;
    #pragma unroll
    for (int c = 0; c < 4; ++c) {
        const int j4 = c * 32 + lane;   // v4f index, coalesced per wave
        const v4f cv = __builtin_nontemporal_load(cp + j4);  // stream once, TH=NT
        const v4f vv = v4[j4];
        const v4f cn = fv * cv + s1 * vv;
        __builtin_nontemporal_store(cn, co + j4);            // written once, TH=NT
        const v4f qv = q4[j4];
        dot += cn.x * qv.x + cn.y * qv.y + cn.z * qv.z + cn.w * qv.w;
    }

    // wave32 reduction
    for (int off = 16; off; off >>= 1)
        dot += __shfl_down(dot, off, 32);
    if (lane == 0) num[gidx] = dot;
}

// ---------------------------------------------------------------------------
// Kernel 4: h_tilde = num/denom, GroupNorm(1 group) over H, output gate.
// One block per batch row.
// ---------------------------------------------------------------------------
__global__ __launch_bounds__(256) void norm_kernel(
    const float* __restrict__ num, const float* __restrict__ denom,
    const float* __restrict__ ob, const float* __restrict__ gamma,
    const float* __restrict__ beta, float* __restrict__ h_out)
{
    __shared__ float rs[256];
    __shared__ float rs2[256];
    const int b = blockIdx.x, t = threadIdx.x;
    const int i0 = b * Hsz + t, i1 = i0 + 256;

    const float h0 = num[i0] / denom[i0];
    const float h1 = num[i1] / denom[i1];
    rs[t]  = h0 + h1;
    rs2[t] = h0 * h0 + h1 * h1;
    __syncthreads();
    for (int off = 128; off; off >>= 1) {
        if (t < off) { rs[t] += rs[t + off]; rs2[t] += rs2[t + off]; }
        __syncthreads();
    }
    const float mu   = rs[0] * (1.f / (float)Hsz);
    const float var  = rs2[0] * (1.f / (float)Hsz) - mu * mu;
    const float rstd = rsqrtf(var + 1e-5f);

    h_out[i0] = ob[i0] * (gamma[t]       * (h0 - mu) * rstd + beta[t]);
    h_out[i1] = ob[i1] * (gamma[t + 256] * (h1 - mu) * rstd + beta[t + 256]);
}

// ---------------------------------------------------------------------------
extern "C" void kernel_launch(void* const* d_in, const int* in_sizes, int n_in,
                              void* d_out, int out_size, void* d_ws, size_t ws_size,
                              hipStream_t stream) {
    const float* x      = (const float*)d_in[0];
    const float* C_prev = (const float*)d_in[1];
    const float* n_prev = (const float*)d_in[2];
    const float* m_prev = (const float*)d_in[3];
    const float* Wq     = (const float*)d_in[4];
    const float* Wk     = (const float*)d_in[5];
    const float* Wv     = (const float*)d_in[6];
    const float* Wi     = (const float*)d_in[7];
    const float* bi     = (const float*)d_in[8];
    const float* Wf     = (const float*)d_in[9];
    const float* bf     = (const float*)d_in[10];
    const float* Wo     = (const float*)d_in[11];
    const float* bo     = (const float*)d_in[12];
    const float* gamma  = (const float*)d_in[13];
    const float* beta   = (const float*)d_in[14];

    // outputs concatenated in return order: h, C, n, m
    float* out   = (float*)d_out;
    float* h_out = out;
    float* C_out = out + (size_t)Bsz * Hsz;
    float* n_out = C_out + (size_t)Bsz * Hsz * Hsz;
    float* m_out = n_out + (size_t)Bsz * Hsz;

    // workspace: q,k,v,i,f,o,num,denom  (8 * B*H floats = 4 MB)
    float* ws = (float*)d_ws;
    const size_t off = (size_t)Bsz * Hsz;
    float* qb   = ws + 0 * off;
    float* kb   = ws + 1 * off;
    float* vb   = ws + 2 * off;
    float* ib   = ws + 3 * off;
    float* fb   = ws + 4 * off;
    float* ob   = ws + 5 * off;
    float* numb = ws + 6 * off;
    float* denb = ws + 7 * off;

    proj_kernel<<<384, 256, 0, stream>>>(x, Wq, Wk, Wv, Wi, Wf, Wo, bi, bf, bo,
                                         qb, kb, vb, ib, fb, ob);
    gates_kernel<<<256, 256, 0, stream>>>(qb, kb, ib, fb, n_prev, m_prev,
                                          n_out, m_out, denb);
    cell_kernel<<<Bsz * (Hsz / 8), 256, 0, stream>>>(C_prev, vb, qb, ib, kb, fb,
                                                     C_out, numb);
    norm_kernel<<<256, 256, 0, stream>>>(numb, denb, ob, gamma, beta, h_out);
}